// PiecewiseLinearActivation_3831110828573
// MI455X (gfx1250) — compile-verified
//
#include <hip/hip_runtime.h>
#include <cstdint>

#define K_KNOTS 31
#define S_SLOPES 32
#define F_TILE 128
#define ROWS_PER_BLOCK 64
#define TPB 256

typedef int v4i __attribute__((ext_vector_type(4)));
typedef __attribute__((address_space(1))) v4i GV4;   // global int4
typedef __attribute__((address_space(3))) v4i LV4;   // LDS int4

#if defined(__has_builtin)
#if __has_builtin(__builtin_amdgcn_global_load_async_to_lds_b128)
#define HAVE_ASYNC_LDS 1
#endif
#if __has_builtin(__builtin_amdgcn_s_wait_asynccnt)
#define HAVE_WAIT_ASYNC 1
#endif
#endif

__device__ __forceinline__ float softplus_eps(float x) {
    // stable softplus: max(x,0) + log1p(exp(-|x|)), plus slope floor eps
    return fmaxf(x, 0.0f) + log1pf(expf(-fabsf(x))) + 1e-3f;
}

// ---------------------------------------------------------------------------
// Kernel 1: build per-feature tables into d_ws, transposed tiled layout:
//   ws[region][tile][k][128]  with region stride = F*32 floats,
//   regions: 0 = sorted knots xs, 1 = slope_c, 2 = knot y values
// ---------------------------------------------------------------------------
__global__ void pwl_build_tables(const float* __restrict__ x_pos,
                                 const float* __restrict__ slope,
                                 const float* __restrict__ y_bias,
                                 float* __restrict__ ws, int F) {
    int f = blockIdx.x * blockDim.x + threadIdx.x;
    if (f >= F) return;

    float a[K_KNOTS];
    #pragma unroll
    for (int k = 0; k < K_KNOTS; ++k) a[k] = x_pos[f * K_KNOTS + k];

    // insertion sort (31 elements, per-thread)
    for (int i = 1; i < K_KNOTS; ++i) {
        float key = a[i];
        int j = i - 1;
        while (j >= 0 && a[j] > key) { a[j + 1] = a[j]; --j; }
        a[j + 1] = key;
    }

    float sc[S_SLOPES];
    #pragma unroll
    for (int j = 0; j < S_SLOPES; ++j) sc[j] = softplus_eps(slope[f * S_SLOPES + j]);

    float y[K_KNOTS];
    y[0] = a[0] + y_bias[f];
    #pragma unroll
    for (int k = 1; k < K_KNOTS; ++k) y[k] = y[k - 1] + (a[k] - a[k - 1]) * sc[k];

    const int tile = f / F_TILE;
    const int fl = f % F_TILE;
    const int rstride = F * 32;  // floats per region
    float* xs_ws = ws + tile * (32 * F_TILE) + fl;
    float* sl_ws = xs_ws + rstride;
    float* y_ws  = xs_ws + 2 * rstride;

    #pragma unroll
    for (int k = 0; k < K_KNOTS; ++k) xs_ws[k * F_TILE] = a[k];
    xs_ws[31 * F_TILE] = __int_as_float(0x7f800000);  // +inf pad (unused)
    #pragma unroll
    for (int j = 0; j < S_SLOPES; ++j) sl_ws[j * F_TILE] = sc[j];
    #pragma unroll
    for (int k = 0; k < K_KNOTS; ++k) y_ws[k * F_TILE] = y[k];
    y_ws[31 * F_TILE] = 0.0f;  // pad (unused)
}

// ---------------------------------------------------------------------------
// Kernel 2: main streaming kernel. Block = 128-feature tile x 64 rows.
// Tables staged to LDS with gfx1250 async global->LDS DMA.
// LDS layout [k][128] => per-lane table gathers are bank-conflict-free.
// ---------------------------------------------------------------------------
__global__ __launch_bounds__(TPB) void pwl_apply(const float* __restrict__ inp,
                                                 const float* __restrict__ ws,
                                                 float* __restrict__ out,
                                                 float* __restrict__ sout,
                                                 int F) {
    __shared__ float smem[3 * 32 * F_TILE];  // 12288 floats = 48 KB

    const int tid = threadIdx.x;
    const int ftile = blockIdx.x;
    const long long rowbase = (long long)blockIdx.y * ROWS_PER_BLOCK;
    const int rstride = F * 32;

    const float* gbase = ws + ftile * (32 * F_TILE);

#if HAVE_ASYNC_LDS
    // 3 regions x 4096 floats = 3072 float4 transfers; 12 per thread
    for (int i = tid; i < 3 * 1024; i += TPB) {
        const int reg = i >> 10;
        const int off = i & 1023;
        const float* src = gbase + reg * rstride + off * 4;
        float* dst = smem + reg * 4096 + off * 4;
        __builtin_amdgcn_global_load_async_to_lds_b128(
            (GV4*)(uintptr_t)src, (LV4*)(uintptr_t)dst, 0, 0);
    }
#if HAVE_WAIT_ASYNC
    __builtin_amdgcn_s_wait_asynccnt(0);
#else
    asm volatile("s_wait_asynccnt 0" ::: "memory");
#endif
    __syncthreads();
#else
    for (int i = tid; i < 3 * 1024; i += TPB) {
        const int reg = i >> 10;
        const int off = i & 1023;
        *(float4*)(smem + reg * 4096 + off * 4) =
            *(const float4*)(gbase + reg * rstride + off * 4);
    }
    __syncthreads();
#endif

    const float* xs_s = smem;           // [32][128], rows 0..30 valid
    const float* sl_s = smem + 4096;    // [32][128]
    const float* y_s  = smem + 8192;    // [32][128], rows 0..30 valid

    const int fl = tid & (F_TILE - 1);
    const int tr = tid >> 7;  // 0..1
    const int f = ftile * F_TILE + fl;

    #pragma unroll 4
    for (int rr = tr; rr < ROWS_PER_BLOCK; rr += 2) {
        const long long gi = (rowbase + rr) * (long long)F + f;
        const float v = __builtin_nontemporal_load(inp + gi);

        // searchsorted(side='right') over 31 sorted knots: idx = #(xs <= v) in [0,31]
        int idx = 0;
        #pragma unroll
        for (int s = 16; s > 0; s >>= 1) {
            const int t = idx + s;                  // t-1 in [0,30] always
            if (xs_s[(t - 1) * F_TILE + fl] <= v) idx = t;
        }

        const float sl = sl_s[idx * F_TILE + fl];
        const int xi = (idx > 0) ? (idx - 1) : 0;
        const float xk = xs_s[xi * F_TILE + fl];
        const float yk = y_s[xi * F_TILE + fl];
        const float o = fmaf(v - xk, sl, yk);

        __builtin_nontemporal_store(o, out + gi);
        __builtin_nontemporal_store(sl, sout + gi);
    }
}

// ---------------------------------------------------------------------------
extern "C" void kernel_launch(void* const* d_in, const int* in_sizes, int n_in,
                              void* d_out, int out_size, void* d_ws, size_t ws_size,
                              hipStream_t stream) {
    const float* inputs = (const float*)d_in[0];  // [B, F]
    const float* x_pos  = (const float*)d_in[1];  // [F, 31]
    const float* slope  = (const float*)d_in[2];  // [F, 32]
    const float* y_bias = (const float*)d_in[3];  // [F]

    const int F = in_sizes[1] / K_KNOTS;                    // 512
    const long long B = (long long)in_sizes[0] / F;         // 131072

    float* out  = (float*)d_out;
    float* sout = out + (long long)B * F;
    float* ws   = (float*)d_ws;  // needs 3*F*32*4 = 192 KB

    pwl_build_tables<<<(F + 255) / 256, 256, 0, stream>>>(x_pos, slope, y_bias, ws, F);

    dim3 grid(F / F_TILE, (unsigned)(B / ROWS_PER_BLOCK));
    pwl_apply<<<grid, TPB, 0, stream>>>(inputs, ws, out, sout, F);
}